// HyperAggModel_52467320487984
// MI455X (gfx1250) — compile-verified
//
#include <hip/hip_runtime.h>

// Problem constants (match reference)
#define NN     100000   // nodes
#define NHE    50000    // hyperedges
#define EE     500000   // incidences
#define DDIM   128      // embed dim
#define RREL   500      // relations
#define NDIS7  7        // distance rows
#define BQ     512      // queries / positives
#define NNEG8  8192     // negatives

typedef __attribute__((ext_vector_type(2))) float v2f;
typedef __attribute__((ext_vector_type(8))) float v8f;

// ---------------------------------------------------------------------------
// Count incidences per hyperedge / node (once; identical across layers)
// ---------------------------------------------------------------------------
__global__ __launch_bounds__(256) void count_kernel(
    const int* __restrict__ node_idx, const int* __restrict__ hedge_idx,
    float* __restrict__ cnt_h, float* __restrict__ cnt_n)
{
    int e = blockIdx.x * blockDim.x + threadIdx.x;
    if (e >= EE) return;
    atomicAdd(cnt_h + hedge_idx[e], 1.0f);
    atomicAdd(cnt_n + node_idx[e], 1.0f);
}

// counts -> inverse mean factors. Hedge segments see each incidence twice (2E list).
__global__ __launch_bounds__(256) void finalize_inv_kernel(
    float* __restrict__ inv_h, float* __restrict__ inv_n)
{
    int i = blockIdx.x * blockDim.x + threadIdx.x;
    if (i < NHE) {
        float c = 2.0f * inv_h[i];
        inv_h[i] = 1.0f / fmaxf(c, 1.0f);
    } else if (i < NHE + NN) {
        int j = i - NHE;
        float c = inv_n[j];
        inv_n[j] = 1.0f / fmaxf(c, 1.0f);
    }
}

// ---------------------------------------------------------------------------
// Scatter kernels: 32 threads per edge, each handles 4 consecutive dwords.
// V2E contributions for one edge share the same target row, so the two
// gathers (entity/distance row + relation row) are fused before the atomic.
// ---------------------------------------------------------------------------
__global__ __launch_bounds__(256) void scatter_v2e_k0(
    const int* __restrict__ node_idx, const int* __restrict__ hedge_idx,
    const int* __restrict__ edge_attr, const int* __restrict__ v_dis,
    const float* __restrict__ dis_emb, const float* __restrict__ rel_emb,
    float* __restrict__ hedge_sum)
{
    unsigned t = blockIdx.x * blockDim.x + threadIdx.x;
    int e = (int)(t >> 5);
    if (e >= EE) return;
    int c = (int)(t & 31) * 4;
    int h = hedge_idx[e];
    int dr = v_dis[node_idx[e]];
    int rr = edge_attr[e];
    const float4 dv = *(const float4*)(dis_emb + dr * DDIM + c);
    const float4 rv = *(const float4*)(rel_emb + rr * DDIM + c);
    float* dst = hedge_sum + (size_t)h * DDIM + c;
    atomicAdd(dst + 0, dv.x + rv.x);
    atomicAdd(dst + 1, dv.y + rv.y);
    atomicAdd(dst + 2, dv.z + rv.z);
    atomicAdd(dst + 3, dv.w + rv.w);
}

__global__ __launch_bounds__(256) void scatter_v2e_ent(
    const int* __restrict__ node_idx, const int* __restrict__ hedge_idx,
    const int* __restrict__ edge_attr, const float* __restrict__ ent_emb,
    const float* __restrict__ rel_emb, float* __restrict__ hedge_sum)
{
    unsigned t = blockIdx.x * blockDim.x + threadIdx.x;
    int e = (int)(t >> 5);
    if (e >= EE) return;
    int c = (int)(t & 31) * 4;
    int h = hedge_idx[e];
    int v = node_idx[e];
    int rr = edge_attr[e];
    const float4 ev = *(const float4*)(ent_emb + (size_t)v * DDIM + c);
    const float4 rv = *(const float4*)(rel_emb + rr * DDIM + c);
    float* dst = hedge_sum + (size_t)h * DDIM + c;
    atomicAdd(dst + 0, ev.x + rv.x);
    atomicAdd(dst + 1, ev.y + rv.y);
    atomicAdd(dst + 2, ev.z + rv.z);
    atomicAdd(dst + 3, ev.w + rv.w);
}

__global__ __launch_bounds__(256) void scatter_e2v(
    const int* __restrict__ node_idx, const int* __restrict__ hedge_idx,
    const float* __restrict__ hedge_emb, float* __restrict__ ent_sum)
{
    unsigned t = blockIdx.x * blockDim.x + threadIdx.x;
    int e = (int)(t >> 5);
    if (e >= EE) return;
    int c = (int)(t & 31) * 4;
    int h = hedge_idx[e];
    int v = node_idx[e];
    const float4 hv = *(const float4*)(hedge_emb + (size_t)h * DDIM + c);
    float* dst = ent_sum + (size_t)v * DDIM + c;
    atomicAdd(dst + 0, hv.x);
    atomicAdd(dst + 1, hv.y);
    atomicAdd(dst + 2, hv.z);
    atomicAdd(dst + 3, hv.w);
}

// ---------------------------------------------------------------------------
// Fused mean + dual GEMM + ReLU via V_WMMA_F32_16X16X4_F32.
//   Out[r,:] = relu( (Asum[r,:]*inv[r]) @ Wsrc  +  Aself[r,:] @ Wself )
// Block = 256 threads (8 waves) -> 16 rows x 128 cols; wave w owns cols
// [16w,16w+16). In-place (Out==Aself) is safe: tile r only reads rows of r.
// ---------------------------------------------------------------------------
__global__ __launch_bounds__(256) void gemm_relu_wmma(
    const float* __restrict__ Asum, const float* __restrict__ inv_cnt,
    const float* __restrict__ Wsrc, const float* __restrict__ Aself,
    const float* __restrict__ Wself, float* __restrict__ Out)
{
    __shared__ float lds_a[16][132];   // stride 132: conflict-free A fetch
    __shared__ float lds_s[16][132];

    const int r0  = blockIdx.x * 16;
    const int tid = threadIdx.x;

    for (int i = tid; i < 16 * DDIM; i += 256) {
        int row = i >> 7, col = i & 127;
        lds_a[row][col] = Asum[(size_t)(r0 + row) * DDIM + col] * inv_cnt[r0 + row];
    }
    if (Aself) {
        for (int i = tid; i < 16 * DDIM; i += 256) {
            int row = i >> 7, col = i & 127;
            lds_s[row][col] = Aself[(size_t)(r0 + row) * DDIM + col];
        }
    }
    __syncthreads();

    const int wave  = tid >> 5;
    const int lane  = tid & 31;
    const int n0    = wave * 16;
    const int m     = lane & 15;      // A row / B-D column within tile
    const int hh    = lane >> 4;      // lane half
    const int kbase = hh * 2;         // K sub-offset per ISA A/B layout

    v8f acc = {};
    #pragma unroll 8
    for (int kk = 0; kk < 32; ++kk) {
        int k0 = kk * 4;
        v2f a, b;
        a.x = lds_a[m][k0 + kbase];
        a.y = lds_a[m][k0 + kbase + 1];
        b.x = Wsrc[(k0 + kbase) * DDIM + n0 + m];
        b.y = Wsrc[(k0 + kbase + 1) * DDIM + n0 + m];
        acc = __builtin_amdgcn_wmma_f32_16x16x4_f32(false, a, false, b,
                                                    (short)0, acc, false, false);
    }
    if (Aself) {
        #pragma unroll 8
        for (int kk = 0; kk < 32; ++kk) {
            int k0 = kk * 4;
            v2f a, b;
            a.x = lds_s[m][k0 + kbase];
            a.y = lds_s[m][k0 + kbase + 1];
            b.x = Wself[(k0 + kbase) * DDIM + n0 + m];
            b.y = Wself[(k0 + kbase + 1) * DDIM + n0 + m];
            acc = __builtin_amdgcn_wmma_f32_16x16x4_f32(false, a, false, b,
                                                        (short)0, acc, false, false);
        }
    }
    // D layout: VGPR v -> row v + 8*half, col = lane&15 (within tile)
    #pragma unroll
    for (int v = 0; v < 8; ++v) {
        float val = acc[v];
        val = val > 0.0f ? val : 0.0f;
        Out[(size_t)(r0 + v + hh * 8) * DDIM + n0 + m] = val;
    }
}

// ---------------------------------------------------------------------------
// Final output gather: (h_out[512], pos[512], neg[8192]) x 128, concatenated.
// ---------------------------------------------------------------------------
__global__ __launch_bounds__(256) void gather_out_kernel(
    const float* __restrict__ hedge_emb, const float* __restrict__ ent_emb,
    const int* __restrict__ src_ids, const int* __restrict__ pos_ids,
    const int* __restrict__ neg_ids, float* __restrict__ out)
{
    unsigned t = blockIdx.x * blockDim.x + threadIdx.x;
    int row = (int)(t >> 5);
    const int TOT = BQ + BQ + NNEG8;
    if (row >= TOT) return;
    int c = (int)(t & 31) * 4;
    const float* src;
    if (row < BQ)            src = hedge_emb + (size_t)src_ids[row] * DDIM;
    else if (row < 2 * BQ)   src = ent_emb + (size_t)pos_ids[row - BQ] * DDIM;
    else                     src = ent_emb + (size_t)neg_ids[row - 2 * BQ] * DDIM;
    *(float4*)(out + (size_t)row * DDIM + c) = *(const float4*)(src + c);
}

// ---------------------------------------------------------------------------
extern "C" void kernel_launch(void* const* d_in, const int* in_sizes, int n_in,
                              void* d_out, int out_size, void* d_ws, size_t ws_size,
                              hipStream_t stream)
{
    const int*   node_idx   = (const int*)d_in[0];
    const int*   hedge_idx  = (const int*)d_in[1];
    const int*   edge_attr  = (const int*)d_in[2];
    const int*   v_dis      = (const int*)d_in[3];
    const int*   src_ids    = (const int*)d_in[4];
    const int*   pos_ids    = (const int*)d_in[5];
    const int*   neg_ids    = (const int*)d_in[6];
    const float* dis_emb    = (const float*)d_in[7];
    const float* rel_emb    = (const float*)d_in[8];
    const float* W_v2e_src  = (const float*)d_in[9];
    const float* W_v2e_self = (const float*)d_in[10];
    const float* W_e2v_src  = (const float*)d_in[11];
    const float* W_e2v_self = (const float*)d_in[12];
    float* out = (float*)d_out;

    const size_t DDmat = (size_t)DDIM * DDIM;

    // Workspace layout (floats)
    float* ws = (float*)d_ws;
    float* hedge_sum = ws;                    // NHE*DDIM
    float* ent_sum   = hedge_sum + (size_t)NHE * DDIM;  // NN*DDIM
    float* hedge_emb = ent_sum + (size_t)NN * DDIM;     // NHE*DDIM
    float* ent_emb   = hedge_emb + (size_t)NHE * DDIM;  // NN*DDIM
    float* inv_h     = ent_emb + (size_t)NN * DDIM;     // NHE
    float* inv_n     = inv_h + NHE;                     // NN
    size_t need = ((size_t)(NHE + NN) * DDIM * 2 + NHE + NN) * sizeof(float);
    if (ws_size < need) return;

    const dim3 blk(256);
    const dim3 grid_scatter((EE * 32) / 256);          // 62500
    const dim3 grid_cnt((EE + 255) / 256);
    const dim3 grid_inv((NHE + NN + 255) / 256);
    const dim3 grid_gemm_h(NHE / 16);                  // 3125
    const dim3 grid_gemm_n(NN / 16);                   // 6250
    const dim3 grid_gather(((BQ + BQ + NNEG8) * 32) / 256);

    // ---- counts (same for every layer) ----
    hipMemsetAsync(inv_h, 0, (size_t)(NHE + NN) * sizeof(float), stream);
    count_kernel<<<grid_cnt, blk, 0, stream>>>(node_idx, hedge_idx, inv_h, inv_n);
    finalize_inv_kernel<<<grid_inv, blk, 0, stream>>>(inv_h, inv_n);

    // ---- k = 0 : V2E (distance + relation), no self ----
    hipMemsetAsync(hedge_sum, 0, (size_t)NHE * DDIM * sizeof(float), stream);
    scatter_v2e_k0<<<grid_scatter, blk, 0, stream>>>(node_idx, hedge_idx, edge_attr,
                                                     v_dis, dis_emb, rel_emb, hedge_sum);
    gemm_relu_wmma<<<grid_gemm_h, blk, 0, stream>>>(hedge_sum, inv_h, W_v2e_src,
                                                    nullptr, nullptr, hedge_emb);

    // ---- k = 0 : E2V, no self ----
    hipMemsetAsync(ent_sum, 0, (size_t)NN * DDIM * sizeof(float), stream);
    scatter_e2v<<<grid_scatter, blk, 0, stream>>>(node_idx, hedge_idx, hedge_emb, ent_sum);
    gemm_relu_wmma<<<grid_gemm_n, blk, 0, stream>>>(ent_sum, inv_n, W_e2v_src,
                                                    nullptr, nullptr, ent_emb);

    // ---- k = 1 : V2E (entity + relation) with self ----
    hipMemsetAsync(hedge_sum, 0, (size_t)NHE * DDIM * sizeof(float), stream);
    scatter_v2e_ent<<<grid_scatter, blk, 0, stream>>>(node_idx, hedge_idx, edge_attr,
                                                      ent_emb, rel_emb, hedge_sum);
    gemm_relu_wmma<<<grid_gemm_h, blk, 0, stream>>>(hedge_sum, inv_h, W_v2e_src + DDmat,
                                                    hedge_emb, W_v2e_self + DDmat, hedge_emb);

    // ---- k = 1 : E2V with self ----
    hipMemsetAsync(ent_sum, 0, (size_t)NN * DDIM * sizeof(float), stream);
    scatter_e2v<<<grid_scatter, blk, 0, stream>>>(node_idx, hedge_idx, hedge_emb, ent_sum);
    gemm_relu_wmma<<<grid_gemm_n, blk, 0, stream>>>(ent_sum, inv_n, W_e2v_src + DDmat,
                                                    ent_emb, W_e2v_self + DDmat, ent_emb);

    // ---- k = 2 : V2E with self (final hedge layer) ----
    hipMemsetAsync(hedge_sum, 0, (size_t)NHE * DDIM * sizeof(float), stream);
    scatter_v2e_ent<<<grid_scatter, blk, 0, stream>>>(node_idx, hedge_idx, edge_attr,
                                                      ent_emb, rel_emb, hedge_sum);
    gemm_relu_wmma<<<grid_gemm_h, blk, 0, stream>>>(hedge_sum, inv_h, W_v2e_src + 2 * DDmat,
                                                    hedge_emb, W_v2e_self + 2 * DDmat, hedge_emb);

    // ---- outputs ----
    gather_out_kernel<<<grid_gather, blk, 0, stream>>>(hedge_emb, ent_emb,
                                                       src_ids, pos_ids, neg_ids, out);
}